// STAGATEConv_27212912787632
// MI455X (gfx1250) — compile-verified
//
#include <hip/hip_runtime.h>

typedef float v2f __attribute__((ext_vector_type(2)));
typedef float v8f __attribute__((ext_vector_type(8)));

#define N_NODES   50000
#define N_EDGES   800000
#define IN_DIM    128
#define OUT_DIM   64
#define NEG_SLOPE 0.2f
#define EPSV      1e-8f

// ---------------------------------------------------------------------------
// Kernel 0: out[n][d] = bias[d]  (bias folded into init), denom = 0
// ---------------------------------------------------------------------------
__global__ void k_init(float* __restrict__ out, const float* __restrict__ bias,
                       float* __restrict__ denom) {
    int i = blockIdx.x * blockDim.x + threadIdx.x;
    if (i < N_NODES * OUT_DIM) out[i] = bias[i & (OUT_DIM - 1)];
    if (i < N_NODES)           denom[i] = 0.0f;
}

// ---------------------------------------------------------------------------
// Kernel 1: h = x @ W  via V_WMMA_F32_16X16X4_F32 (fp32-exact matrix core)
// One block = 16 output rows; 4 waves each own a 16-col tile of OUT_DIM=64.
// A 16x4 f32 layout: lane m = lane&15, VGPR v holds k = kb + 2*(lane>>4) + v.
// B  4x16 f32 layout: lane n = lane&15, same k striping.
// C/D 16x16 f32: VGPR r -> row = r + 8*(lane>>4), col = lane&15.
// ---------------------------------------------------------------------------
__global__ void k_gemm(const float* __restrict__ x, const float* __restrict__ W,
                       float* __restrict__ h) {
    const int lane = threadIdx.x & 31;
    const int wave = threadIdx.x >> 5;          // 0..3
    const int row_base = blockIdx.x * 16;
    const int col_base = wave * 16;
    const int m  = lane & 15;
    const int kg = lane >> 4;                   // 0 or 1

    int row = row_base + m;
    if (row >= N_NODES) row = N_NODES - 1;      // clamp; masked at store
    const float* xrow = x + (long)row * IN_DIM;

    v8f c = {0.f, 0.f, 0.f, 0.f, 0.f, 0.f, 0.f, 0.f};
#pragma unroll
    for (int kb = 0; kb < IN_DIM; kb += 4) {
        const int k = kb + 2 * kg;
        v2f a, b;
        a.x = xrow[k];
        a.y = xrow[k + 1];
        b.x = W[(long)k       * OUT_DIM + col_base + m];
        b.y = W[(long)(k + 1) * OUT_DIM + col_base + m];
        // (neg_a, A, neg_b, B, c_mod, C, reuse_a, reuse_b)
        c = __builtin_amdgcn_wmma_f32_16x16x4_f32(false, a, false, b,
                                                  (short)0, c, false, false);
    }
#pragma unroll
    for (int r = 0; r < 8; ++r) {
        const int orow = row_base + r + 8 * kg;
        if (orow < N_NODES)
            h[(long)orow * OUT_DIM + col_base + m] = c[r];
    }
}

// ---------------------------------------------------------------------------
// Kernel 2: per-node attention scalars  s_src[n]=h[n]·a_src, s_dst[n]=h[n]·a_dst
// One wave per node; lane covers dims {lane, lane+32}; wave32 xor-reduce.
// ---------------------------------------------------------------------------
__global__ void k_scores(const float* __restrict__ h,
                         const float* __restrict__ att_src,
                         const float* __restrict__ att_dst,
                         float* __restrict__ s_src, float* __restrict__ s_dst) {
    const int lane = threadIdx.x & 31;
    const int wave = threadIdx.x >> 5;
    const int node = blockIdx.x * (blockDim.x >> 5) + wave;
    if (node >= N_NODES) return;
    const float v0 = h[(long)node * OUT_DIM + lane];
    const float v1 = h[(long)node * OUT_DIM + lane + 32];
    float ps = v0 * att_src[lane] + v1 * att_src[lane + 32];
    float pd = v0 * att_dst[lane] + v1 * att_dst[lane + 32];
#pragma unroll
    for (int off = 16; off > 0; off >>= 1) {
        ps += __shfl_xor(ps, off, 32);
        pd += __shfl_xor(pd, off, 32);
    }
    if (lane == 0) { s_src[node] = ps; s_dst[node] = pd; }
}

// ---------------------------------------------------------------------------
// Kernel 3: per-edge alpha = sigmoid(leaky_relu(s_src[s]+s_dst[d]));
//           denom[d] += alpha  (L2-resident atomics)
// ---------------------------------------------------------------------------
__global__ void k_edge(const int* __restrict__ ei,
                       const float* __restrict__ s_src,
                       const float* __restrict__ s_dst,
                       float* __restrict__ alpha, float* __restrict__ denom) {
    const int e = blockIdx.x * blockDim.x + threadIdx.x;
    if (e >= N_EDGES) return;
    const int s = ei[e];
    const int d = ei[N_EDGES + e];
    float v = s_src[s] + s_dst[d];
    v = (v > 0.f) ? v : NEG_SLOPE * v;
    const float a = 1.f / (1.f + expf(-v));
    alpha[e] = a;
    atomicAdd(&denom[d], a);
}

// ---------------------------------------------------------------------------
// Kernel 4: scatter  out[d] += (alpha/denom[d]) * h[s]
// One wave per edge iteration; lanes cover dims {lane, lane+32}.
// ---------------------------------------------------------------------------
__global__ void k_scatter(const int* __restrict__ ei,
                          const float* __restrict__ h,
                          const float* __restrict__ alpha,
                          const float* __restrict__ denom,
                          float* __restrict__ out) {
    const int lane = threadIdx.x & 31;
    const int wave = threadIdx.x >> 5;
    const int wavesTotal = gridDim.x * (blockDim.x >> 5);
    for (int e = blockIdx.x * (blockDim.x >> 5) + wave; e < N_EDGES; e += wavesTotal) {
        const int s = ei[e];
        const int d = ei[N_EDGES + e];
        const float an = alpha[e] / (denom[d] + EPSV);
        const float v0 = h[(long)s * OUT_DIM + lane]      * an;
        const float v1 = h[(long)s * OUT_DIM + lane + 32] * an;
        atomicAdd(&out[(long)d * OUT_DIM + lane],      v0);
        atomicAdd(&out[(long)d * OUT_DIM + lane + 32], v1);
    }
}

// ---------------------------------------------------------------------------
extern "C" void kernel_launch(void* const* d_in, const int* in_sizes, int n_in,
                              void* d_out, int out_size, void* d_ws, size_t ws_size,
                              hipStream_t stream) {
    const float* x       = (const float*)d_in[0];   // (N, 128)
    const int*   ei      = (const int*)  d_in[1];   // (2, E)
    const float* W       = (const float*)d_in[2];   // (128, 64)
    const float* att_src = (const float*)d_in[3];   // (64,)
    const float* att_dst = (const float*)d_in[4];   // (64,)
    const float* bias    = (const float*)d_in[5];   // (64,)
    float* out = (float*)d_out;                     // (N, 64)

    float* ws    = (float*)d_ws;
    float* h     = ws;                               // N*64 = 3.2M floats
    float* s_src = h + (size_t)N_NODES * OUT_DIM;    // N
    float* s_dst = s_src + N_NODES;                  // N
    float* denom = s_dst + N_NODES;                  // N
    float* alpha = denom + N_NODES;                  // E

    k_init   <<<(N_NODES * OUT_DIM + 255) / 256, 256, 0, stream>>>(out, bias, denom);
    k_gemm   <<<(N_NODES + 15) / 16, 128, 0, stream>>>(x, W, h);
    k_scores <<<(N_NODES + 7) / 8, 256, 0, stream>>>(h, att_src, att_dst, s_src, s_dst);
    k_edge   <<<(N_EDGES + 255) / 256, 256, 0, stream>>>(ei, s_src, s_dst, alpha, denom);
    k_scatter<<<12500, 256, 0, stream>>>(ei, h, alpha, denom, out);
}